// TensorChannel_90984587198691
// MI455X (gfx1250) — compile-verified
//
#include <hip/hip_runtime.h>
#include <hip/hip_bf16.h>

typedef __attribute__((ext_vector_type(2))) float v2f;
typedef __attribute__((ext_vector_type(8))) float v8f;

// --------------------------------------------------------------------------
// WMMA f32 16x16x4 fragment conventions (CDNA5 ISA):
//   A (16x4, MxK): lanes 0-15 hold M=lane, {K0,K1}; lanes 16-31 hold M=lane-16, {K2,K3}
//   B (4x16, KxN): lanes 0-15 hold N=lane, {K0,K1}; lanes 16-31 hold N=lane-16, {K2,K3}
//   C/D (16x16):   VGPR r: lanes0-15 -> (M=r, N=lane); lanes16-31 -> (M=r+8, N=lane-16)
// --------------------------------------------------------------------------

__device__ __forceinline__ float fast_sigmoid(float x) {
    return 1.0f / (1.0f + __expf(-x));
}

// Async 16-byte copy global -> LDS (GLOBAL_LOAD_ASYNC_TO_LDS_B128, GV mode).
// ldsByteAddr: workgroup-relative LDS byte address (generic ptr truncated to 32b,
// per ISA 10.2).  gaddr: 64-bit global VA.
__device__ __forceinline__ void async_copy_b128(uint32_t ldsByteAddr, uint64_t gaddr) {
    asm volatile("global_load_async_to_lds_b128 %0, %1, off"
                 :: "v"(ldsByteAddr), "v"(gaddr) : "memory");
}
__device__ __forceinline__ void wait_async_zero() {
    asm volatile("s_wait_asynccnt 0x0" ::: "memory");
}

// ==========================================================================
// Kernel 1: tensor_init.  16 nodes per workgroup, 256 threads (8 wave32).
//   t = silu(LN(h @ W1^T + b1)); t = silu(LN(t @ W2^T + b2)) -> tensors[N,144]
// ==========================================================================
__global__ __launch_bounds__(256) void node_init_kernel(
    const float* __restrict__ h,
    const float* __restrict__ W1, const float* __restrict__ b1,
    const float* __restrict__ g1, const float* __restrict__ be1,
    const float* __restrict__ W2, const float* __restrict__ b2,
    const float* __restrict__ g2, const float* __restrict__ be2,
    float* __restrict__ tensors, int N)
{
    __shared__ float sH[16 * 132];   // h tile, stride 132 (row = 528B, 16B aligned)
    __shared__ float sA[16 * 146];   // y1 -> t1, stride 146
    __shared__ float sB[16 * 146];   // y2
    __shared__ float sRs[256], sRq[256];

    const int tid = threadIdx.x;
    const int n0  = blockIdx.x * 16;

    // stage h tile via async b128: 16 nodes x 32 segments = 512, 2 per thread
    for (int s = tid; s < 512; s += 256) {
        int m = s >> 5;                      // node slot
        int j = (s & 31) * 4;                // float offset in row
        int node = n0 + m; if (node >= N) node = N - 1;
        uint32_t lva = (uint32_t)(uintptr_t)&sH[m * 132 + j];
        uint64_t ga  = (uint64_t)(uintptr_t)(h + (size_t)node * 128 + j);
        async_copy_b128(lva, ga);
    }
    wait_async_zero();
    __syncthreads();

    const int lane = tid & 31;
    const int wave = tid >> 5;
    const int mrow = lane & 15;
    const int koff = (lane >> 4) * 2;
    const int mbase = (lane >> 4) * 8;

    // ---- GEMM1: [16x128] @ W1^T[128x144], 9 N-tiles over 8 waves ----
    for (int nt = wave; nt < 9; nt += 8) {
        const int c0 = nt * 16;
        v8f acc = {};
        for (int k = 0; k < 128; k += 4) {
            v2f a = *reinterpret_cast<const v2f*>(&sH[mrow * 132 + k + koff]);
            v2f b = *reinterpret_cast<const v2f*>(&W1[(size_t)(c0 + mrow) * 128 + k + koff]);
            acc = __builtin_amdgcn_wmma_f32_16x16x4_f32(false, a, false, b,
                                                        (short)0, acc, false, false);
        }
        for (int r = 0; r < 8; ++r)
            sA[(mbase + r) * 146 + c0 + mrow] = acc[r] + b1[c0 + mrow];
    }
    __syncthreads();

    // ---- LN + SiLU over 144 features, 16 threads per node ----
    {
        int m = tid >> 4, g = tid & 15;
        float s = 0.f, q = 0.f;
        for (int j = g; j < 144; j += 16) { float v = sA[m * 146 + j]; s += v; q += v * v; }
        sRs[tid] = s; sRq[tid] = q;
        __syncthreads();
        for (int off = 8; off > 0; off >>= 1) {
            if (g < off) { sRs[tid] += sRs[tid + off]; sRq[tid] += sRq[tid + off]; }
            __syncthreads();
        }
        float mu   = sRs[m * 16] * (1.0f / 144.0f);
        float var  = sRq[m * 16] * (1.0f / 144.0f) - mu * mu;
        float rstd = rsqrtf(var + 1e-5f);
        for (int j = g; j < 144; j += 16) {
            float v = (sA[m * 146 + j] - mu) * rstd * g1[j] + be1[j];
            sA[m * 146 + j] = v * fast_sigmoid(v);   // SiLU
        }
    }
    __syncthreads();

    // ---- GEMM2: [16x144] @ W2^T[144x144] ----
    for (int nt = wave; nt < 9; nt += 8) {
        const int c0 = nt * 16;
        v8f acc = {};
        for (int k = 0; k < 144; k += 4) {
            v2f a = *reinterpret_cast<const v2f*>(&sA[mrow * 146 + k + koff]);
            v2f b = *reinterpret_cast<const v2f*>(&W2[(size_t)(c0 + mrow) * 144 + k + koff]);
            acc = __builtin_amdgcn_wmma_f32_16x16x4_f32(false, a, false, b,
                                                        (short)0, acc, false, false);
        }
        for (int r = 0; r < 8; ++r)
            sB[(mbase + r) * 146 + c0 + mrow] = acc[r] + b2[c0 + mrow];
    }
    __syncthreads();

    // ---- LN + SiLU + store to workspace ----
    {
        int m = tid >> 4, g = tid & 15;
        float s = 0.f, q = 0.f;
        for (int j = g; j < 144; j += 16) { float v = sB[m * 146 + j]; s += v; q += v * v; }
        sRs[tid] = s; sRq[tid] = q;
        __syncthreads();
        for (int off = 8; off > 0; off >>= 1) {
            if (g < off) { sRs[tid] += sRs[tid + off]; sRq[tid] += sRq[tid + off]; }
            __syncthreads();
        }
        float mu   = sRs[m * 16] * (1.0f / 144.0f);
        float var  = sRq[m * 16] * (1.0f / 144.0f) - mu * mu;
        float rstd = rsqrtf(var + 1e-5f);
        int node = n0 + m;
        if (node < N) {
            for (int j = g; j < 144; j += 16) {
                float v = (sB[m * 146 + j] - mu) * rstd * g2[j] + be2[j];
                tensors[(size_t)node * 144 + j] = v * fast_sigmoid(v);
            }
        }
    }
}

// ==========================================================================
// Kernel 2: per-edge interaction. 16 edges per workgroup, 256 threads.
//   w   = sigmoid(LN([h_col,h_row] @ Wi^T + bi))           (WMMA f32, K=256)
//   st  = [t_col, R_rel t_row R_rel^T]                     ([32,3,3])
//   out[col] += einsum('kij,kc->cij', st, w.reshape(32,16))
// ==========================================================================
__global__ __launch_bounds__(256) void edge_kernel(
    const float* __restrict__ h, const float* __restrict__ frames,
    const int* __restrict__ ei, const float* __restrict__ tensors,
    const float* __restrict__ Wi, const float* __restrict__ bi,
    const float* __restrict__ gi, const float* __restrict__ bei,
    float* __restrict__ out, int N, int E)
{
    // sY: 16 x 512 (stride 516).  smemB: scalars tile (16x260 = 4160 floats,
    // rows 16B-aligned for async b128) then reused as stacked tensors (4608).
    __shared__ float sY[16 * 516];
    __shared__ float smemB[16 * 32 * 9];
    __shared__ float sR[16 * 12];
    __shared__ int   sCol[16], sRow[16];
    __shared__ float sRs[256], sRq[256];

    const int tid = threadIdx.x;
    const int e0  = blockIdx.x * 16;

    if (tid < 16) {
        int e = e0 + tid; if (e >= E) e = E - 1;
        sRow[tid] = ei[e];       // edge_index[0] (sender)
        sCol[tid] = ei[E + e];   // edge_index[1] (receiver / segment id)
    }
    __syncthreads();

    // ---- Phase 1: async-gather stacked scalars [h_col | h_row] into smemB --
    float* sS = smemB;           // stride 260
    // 16 edges x 64 b128-segments (256 floats) = 1024 segments, 4 per thread
    for (int s = tid; s < 1024; s += 256) {
        int m = s >> 6;                       // edge slot
        int j = (s & 63) * 4;                 // float offset in [0,256)
        int node = (j < 128) ? sCol[m] : sRow[m];
        int jj   = j & 127;
        uint32_t lva = (uint32_t)(uintptr_t)&sS[m * 260 + j];
        uint64_t ga  = (uint64_t)(uintptr_t)(h + (size_t)node * 128 + jj);
        async_copy_b128(lva, ga);
    }
    wait_async_zero();
    __syncthreads();

    // ---- Phase 2: WMMA GEMM Y[16x512] = sS[16x256] @ Wi^T ----
    const int lane = tid & 31;
    const int wave = tid >> 5;
    const int mrow = lane & 15;
    const int koff = (lane >> 4) * 2;
    const int mbase = (lane >> 4) * 8;
    for (int nt = wave; nt < 32; nt += 8) {
        const int c0 = nt * 16;
        v8f acc = {};
        for (int k = 0; k < 256; k += 4) {
            v2f a = *reinterpret_cast<const v2f*>(&sS[mrow * 260 + k + koff]);
            v2f b = *reinterpret_cast<const v2f*>(&Wi[(size_t)(c0 + mrow) * 256 + k + koff]);
            acc = __builtin_amdgcn_wmma_f32_16x16x4_f32(false, a, false, b,
                                                        (short)0, acc, false, false);
        }
        for (int r = 0; r < 8; ++r)
            sY[(mbase + r) * 516 + c0 + mrow] = acc[r] + bi[c0 + mrow];
    }
    __syncthreads();

    // ---- Phase 3: LN + sigmoid over 512 per edge (16 threads / edge) ----
    {
        int m = tid >> 4, g = tid & 15;
        float s = 0.f, q = 0.f;
        for (int j = g; j < 512; j += 16) { float v = sY[m * 516 + j]; s += v; q += v * v; }
        sRs[tid] = s; sRq[tid] = q;
        __syncthreads();
        for (int off = 8; off > 0; off >>= 1) {
            if (g < off) { sRs[tid] += sRs[tid + off]; sRq[tid] += sRq[tid + off]; }
            __syncthreads();
        }
        float mu   = sRs[m * 16] * (1.0f / 512.0f);
        float var  = sRq[m * 16] * (1.0f / 512.0f) - mu * mu;
        float rstd = rsqrtf(var + 1e-5f);
        for (int j = g; j < 512; j += 16) {
            float v = (sY[m * 516 + j] - mu) * rstd * gi[j] + bei[j];
            sY[m * 516 + j] = fast_sigmoid(v);
        }
    }
    __syncthreads();

    // ---- Phase 4a: R_rel = Ri^T @ Rj per edge ----
    if (tid < 16) {
        const float* Ri = &frames[(size_t)sCol[tid] * 9];
        const float* Rj = &frames[(size_t)sRow[tid] * 9];
        float ri[9], rj[9];
        for (int a = 0; a < 9; ++a) { ri[a] = Ri[a]; rj[a] = Rj[a]; }
        for (int i2 = 0; i2 < 3; ++i2)
            for (int j2 = 0; j2 < 3; ++j2) {
                float acc = 0.f;
                for (int k2 = 0; k2 < 3; ++k2) acc += ri[k2 * 3 + i2] * rj[k2 * 3 + j2];
                sR[tid * 12 + i2 * 3 + j2] = acc;
            }
    }
    __syncthreads();

    // ---- Phase 4b: stacked tensors (reuses sS region; sS is dead now) ----
    float* sST = smemB;          // [(e*32+k)*9 + ij]
    for (int p = tid; p < 512; p += 256) {
        int m = p >> 5, k = p & 31;
        float* dst = &sST[(m * 32 + k) * 9];
        if (k < 16) {
            const float* src = &tensors[(size_t)sCol[m] * 144 + k * 9];
            for (int a = 0; a < 9; ++a) dst[a] = src[a];
        } else {
            const float* src = &tensors[(size_t)sRow[m] * 144 + (k - 16) * 9];
            float T[9], Rm[9], M1[9];
            for (int a = 0; a < 9; ++a) { T[a] = src[a]; Rm[a] = sR[m * 12 + a]; }
            for (int i2 = 0; i2 < 3; ++i2)             // M1 = R_rel @ T
                for (int j2 = 0; j2 < 3; ++j2) {
                    float acc = 0.f;
                    for (int k2 = 0; k2 < 3; ++k2) acc += Rm[i2 * 3 + k2] * T[k2 * 3 + j2];
                    M1[i2 * 3 + j2] = acc;
                }
            for (int i2 = 0; i2 < 3; ++i2)             // dst = M1 @ R_rel^T
                for (int j2 = 0; j2 < 3; ++j2) {
                    float acc = 0.f;
                    for (int k2 = 0; k2 < 3; ++k2) acc += M1[i2 * 3 + k2] * Rm[j2 * 3 + k2];
                    dst[i2 * 3 + j2] = acc;
                }
        }
    }
    __syncthreads();

    // ---- Phase 5: channel contraction + scatter-add (thread = (edge,ch)) --
    {
        int m = tid >> 4, c = tid & 15;
        int e = e0 + m;
        if (e < E) {
            float r[9];
            for (int a = 0; a < 9; ++a) r[a] = 0.f;
            for (int k = 0; k < 32; ++k) {
                float wv = sY[m * 516 + k * 16 + c];   // int_weights[e,k,c]
                const float* stp = &sST[(m * 32 + k) * 9];
                for (int a = 0; a < 9; ++a) r[a] += stp[a] * wv;
            }
            float* dst = out + (size_t)sCol[m] * 144 + c * 9;
            for (int a = 0; a < 9; ++a) atomicAdd(dst + a, r[a]);
        }
    }
}

// ==========================================================================
// Kernel 3: in-place symmetrization 0.5*(T + T^T) per (node, channel)
// ==========================================================================
__global__ void sym_kernel(float* __restrict__ out, int total)
{
    int t = blockIdx.x * blockDim.x + threadIdx.x;
    if (t >= total) return;
    float* p = out + (size_t)t * 9;
    float s01 = 0.5f * (p[1] + p[3]);
    float s02 = 0.5f * (p[2] + p[6]);
    float s12 = 0.5f * (p[5] + p[7]);
    p[1] = s01; p[3] = s01;
    p[2] = s02; p[6] = s02;
    p[5] = s12; p[7] = s12;
}

// ==========================================================================
extern "C" void kernel_launch(void* const* d_in, const int* in_sizes, int n_in,
                              void* d_out, int out_size, void* d_ws, size_t ws_size,
                              hipStream_t stream)
{
    // setup_inputs order:
    // 0:x 1:h 2:local_frames 3:edge_index 4:batch
    // 5:W1 6:b1 7:g1 8:be1  9:W2 10:b2 11:g2 12:be2  13:Wi 14:bi 15:gi 16:bei
    const float* h      = (const float*)d_in[1];
    const float* frames = (const float*)d_in[2];
    const int*   ei     = (const int*)d_in[3];
    const float* W1  = (const float*)d_in[5];
    const float* b1  = (const float*)d_in[6];
    const float* g1  = (const float*)d_in[7];
    const float* be1 = (const float*)d_in[8];
    const float* W2  = (const float*)d_in[9];
    const float* b2  = (const float*)d_in[10];
    const float* g2  = (const float*)d_in[11];
    const float* be2 = (const float*)d_in[12];
    const float* Wi  = (const float*)d_in[13];
    const float* bi  = (const float*)d_in[14];
    const float* gi  = (const float*)d_in[15];
    const float* bei = (const float*)d_in[16];

    const int N = in_sizes[1] / 128;
    const int E = in_sizes[3] / 2;

    float* tensors = (float*)d_ws;          // [N,144] fp32 scratch
    float* out     = (float*)d_out;         // [N,16,3,3]

    hipMemsetAsync(out, 0, (size_t)out_size * sizeof(float), stream);

    node_init_kernel<<<(N + 15) / 16, 256, 0, stream>>>(
        h, W1, b1, g1, be1, W2, b2, g2, be2, tensors, N);

    edge_kernel<<<(E + 15) / 16, 256, 0, stream>>>(
        h, frames, ei, tensors, Wi, bi, gi, bei, out, N, E);

    sym_kernel<<<(N * 16 + 255) / 256, 256, 0, stream>>>(out, N * 16);
}